// FullAttnDecodeStep_62680752718086
// MI455X (gfx1250) — compile-verified
//
#include <hip/hip_runtime.h>
#include <hip/hip_bf16.h>
#include <math.h>

#define H 2048
#define NH 16
#define NKV 4
#define HD 128
#define MAX_SEQ 8192
#define RD 32
#define GROUPS 4
#define EPS 1e-6f
#define SCALE 0.08838834764831845f   // 128^-0.5

#define CHUNK 256
#define NCHUNK (MAX_SEQ / CHUNK)
#define PART_STRIDE 520              // 4 m + 4 l + 4*128 acc

// workspace float offsets
#define WS_QG   0        // 4096: per head [q(128) | gate(128)]
#define WS_KP   4096     // 512
#define WS_VP   4608     // 512
#define WS_QF   5120     // 2048: normed+roped q, [h][d]
#define WS_GATT 7168     // 2048: gated attention output, [h][d]
#define WS_PART 9216     // 4*32*520 partials

typedef float v2f __attribute__((ext_vector_type(2)));
typedef float v8f __attribute__((ext_vector_type(8)));

static __device__ __forceinline__ float wave_sum(float v){
  #pragma unroll
  for (int o = 16; o > 0; o >>= 1) v += __shfl_xor(v, o, 32);
  return v;
}
static __device__ __forceinline__ float wave_max(float v){
  #pragma unroll
  for (int o = 16; o > 0; o >>= 1) v = fmaxf(v, __shfl_xor(v, o, 32));
  return v;
}

// ---------------------------------------------------------------------------
// GEMV: y[row] = W[row,:] . x,  C = 2048 fixed. One wave32 per row, float4
// loads (16B/lane/iter), wave shuffle reduction. Pure-BW kernel.
// ---------------------------------------------------------------------------
__global__ __launch_bounds__(256) void gemv2048(const float* __restrict__ W,
                                                const float* __restrict__ x,
                                                float* __restrict__ y, int rows){
  const int lane = threadIdx.x & 31;
  const int wid  = threadIdx.x >> 5;
  const int row  = blockIdx.x * 8 + wid;
  if (row >= rows) return;
  const float4* Wr = (const float4*)(W + (size_t)row * H);
  const float4* xv = (const float4*)x;
  float s = 0.f;
  #pragma unroll
  for (int it = 0; it < 16; ++it){
    float4 a = Wr[lane + 32*it];
    float4 b = xv[lane + 32*it];
    s += a.x*b.x + a.y*b.y + a.z*b.z + a.w*b.w;
  }
  s = wave_sum(s);
  if (lane == 0) y[row] = s;
}

// ---------------------------------------------------------------------------
// Per-head RMSNorm + RoPE; q heads -> ws, kv heads -> new cache row at `pos`.
// Blocks 0..15: q heads. Blocks 16..19: kv heads (also writes v row).
// ---------------------------------------------------------------------------
__global__ __launch_bounds__(128) void norm_rope_write(float* __restrict__ ws,
    const float* __restrict__ qnw, const float* __restrict__ knw,
    const float* __restrict__ cosb, const float* __restrict__ sinb,
    const float* __restrict__ pos,
    float* __restrict__ kc_out, float* __restrict__ vc_out)
{
  __shared__ float sred[4];
  __shared__ float sy[HD];
  const int b = blockIdx.x;
  const int d = threadIdx.x;
  const int lane = d & 31, wid = d >> 5;
  const int P = (int)pos[0];
  float x, wn;
  if (b < NH) { x = ws[WS_QG + b*(2*HD) + d]; wn = qnw[d]; }
  else        { x = ws[WS_KP + (b-NH)*HD + d]; wn = knw[d]; }
  float ps = wave_sum(x*x);
  if (lane == 0) sred[wid] = ps;
  __syncthreads();
  float var = (sred[0]+sred[1]+sred[2]+sred[3]) * (1.0f/HD);
  float y = x * rsqrtf(var + EPS) * (1.0f + wn);
  sy[d] = y;
  __syncthreads();
  float out;
  if (d < 16)      out = y * cosb[d] - sy[d+16] * sinb[d];
  else if (d < RD) out = y * cosb[d] + sy[d-16] * sinb[d];
  else             out = y;
  if (b < NH) {
    ws[WS_QF + b*HD + d] = out;
  } else {
    const int g = b - NH;
    const size_t rowoff = ((size_t)g*MAX_SEQ + P) * HD + d;
    kc_out[rowoff] = out;
    vc_out[rowoff] = ws[WS_VP + g*HD + d];
  }
}

// ---------------------------------------------------------------------------
// Flash-decode partial attention with fp32 WMMA (16x16x4).
// One wave32 per (kv head g, 256-position chunk). A operand = the 4 q heads
// of group g padded to 16 rows. Per 16-pos tile: 32 WMMA for scores (K=128),
// online softmax rescale, then 32 WMMA for weights x V (8 dim-tiles x 4 pos
// groups). Partials (m, l, acc[4][128]) go to workspace.
// ---------------------------------------------------------------------------
__global__ __launch_bounds__(32) void attn_partial(
    const float* __restrict__ qf, const float* __restrict__ Kc,
    const float* __restrict__ Vc, const float* __restrict__ pos,
    float* __restrict__ part)
{
  const int g     = blockIdx.x;
  const int chunk = blockIdx.y;
  const int lane  = threadIdx.x;
  const int r     = lane & 15;     // row / position index within half-wave
  const int pair  = lane >> 4;     // which K-pair (per ISA 16x4 A layout)
  const int P     = (int)pos[0];
  const int base  = chunk * CHUNK;
  float* pb = part + (size_t)(g*NCHUNK + chunk) * PART_STRIDE;

  if (base > P) {   // fully-masked chunk: neutral partial
    for (int j = lane; j < PART_STRIDE; j += 32) pb[j] = (j < 4) ? -1e30f : 0.0f;
    return;
  }

  // A fragments: rows 0..3 = q heads of this group, rows 4..15 zero.
  // Lane l: A[l&15][4k + 2*(l>>4) + {0,1}]  (ISA 7.12.2, 32-bit A 16x4)
  const float* qrow = qf + (size_t)(g*GROUPS + (r & 3)) * HD + 2*pair;
  const float zm = (r < GROUPS) ? 1.0f : 0.0f;
  v2f afrag[32];
  #pragma unroll
  for (int kk = 0; kk < 32; ++kk){
    afrag[kk].x = qrow[4*kk]     * zm;
    afrag[kk].y = qrow[4*kk + 1] * zm;
  }

  v8f vacc[8];
  #pragma unroll
  for (int t8 = 0; t8 < 8; ++t8)
    #pragma unroll
    for (int j = 0; j < 8; ++j) vacc[t8][j] = 0.0f;

  float mcur[4] = {-1e30f,-1e30f,-1e30f,-1e30f};
  float lcur[4] = {0.f,0.f,0.f,0.f};
  const size_t cb = (size_t)g * MAX_SEQ * HD;

  for (int t = 0; t < CHUNK/16; ++t){
    const int s0 = base + 16*t;           // uniform across wave
    if (s0 > P) break;

    // ---- scores tile: D(16x16) = Q(16x128) x K_tile^T, 32 x wmma f32 ----
    v8f c8;
    #pragma unroll
    for (int j = 0; j < 8; ++j) c8[j] = 0.0f;
    // B[k][n] = K[s0+n][4kk+k]; lane l: rows s0+(l&15), cols 4kk+2*pair+{0,1}
    const float* krow = Kc + cb + (size_t)(s0 + r) * HD + 2*pair;
    #pragma unroll
    for (int kk = 0; kk < 32; ++kk){
      v2f bf = *(const v2f*)(krow + 4*kk);
      c8 = __builtin_amdgcn_wmma_f32_16x16x4_f32(false, afrag[kk], false, bf,
                                                 (short)0, c8, false, false);
    }

    // lanes 0..15 of c8[i] hold score[head i][s0+lane]
    const bool vl = (pair == 0) && ((s0 + r) <= P);
    float sc[4], wv[4], sclf[4];
    sc[0] = vl ? c8[0]*SCALE : -1e30f;
    sc[1] = vl ? c8[1]*SCALE : -1e30f;
    sc[2] = vl ? c8[2]*SCALE : -1e30f;
    sc[3] = vl ? c8[3]*SCALE : -1e30f;
    #pragma unroll
    for (int i = 0; i < 4; ++i){
      float tm = wave_max(sc[i]);
      float nm = fmaxf(mcur[i], tm);
      sclf[i] = __expf(mcur[i] - nm);
      wv[i]   = __expf(sc[i] - nm);       // masked lanes -> exactly 0
      float ts = wave_sum(wv[i]);
      lcur[i] = lcur[i]*sclf[i] + ts;
      mcur[i] = nm;
    }
    #pragma unroll
    for (int t8 = 0; t8 < 8; ++t8){
      vacc[t8][0] *= sclf[0]; vacc[t8][1] *= sclf[1];
      vacc[t8][2] *= sclf[2]; vacc[t8][3] *= sclf[3];
    }

    // ---- acc += W_tile(16x16) x V_tile(16x128): 4 pos-groups x 8 dim-tiles
    #pragma unroll
    for (int p = 0; p < 4; ++p){
      const int idx0 = 4*p + 2*pair;      // source lane for weight A-fragment
      float a00 = __shfl(wv[0], idx0, 32), a01 = __shfl(wv[0], idx0+1, 32);
      float a10 = __shfl(wv[1], idx0, 32), a11 = __shfl(wv[1], idx0+1, 32);
      float a20 = __shfl(wv[2], idx0, 32), a21 = __shfl(wv[2], idx0+1, 32);
      float a30 = __shfl(wv[3], idx0, 32), a31 = __shfl(wv[3], idx0+1, 32);
      v2f af;
      af.x = (r==0)?a00:(r==1)?a10:(r==2)?a20:(r==3)?a30:0.0f;
      af.y = (r==0)?a01:(r==1)?a11:(r==2)?a21:(r==3)?a31:0.0f;
      const float* v0 = Vc + cb + (size_t)(s0 + 4*p + 2*pair) * HD;
      const float* v1 = v0 + HD;
      #pragma unroll
      for (int dt = 0; dt < 8; ++dt){
        v2f bf;
        bf.x = v0[16*dt + r];             // B[2*pair+0][16dt+n]
        bf.y = v1[16*dt + r];             // B[2*pair+1][16dt+n]
        vacc[dt] = __builtin_amdgcn_wmma_f32_16x16x4_f32(false, af, false, bf,
                                                         (short)0, vacc[dt],
                                                         false, false);
      }
    }
  }

  if (lane == 0){
    pb[0]=mcur[0]; pb[1]=mcur[1]; pb[2]=mcur[2]; pb[3]=mcur[3];
    pb[4]=lcur[0]; pb[5]=lcur[1]; pb[6]=lcur[2]; pb[7]=lcur[3];
  }
  if (pair == 0){
    #pragma unroll
    for (int dt = 0; dt < 8; ++dt){
      pb[8 + 0*HD + 16*dt + r] = vacc[dt][0];
      pb[8 + 1*HD + 16*dt + r] = vacc[dt][1];
      pb[8 + 2*HD + 16*dt + r] = vacc[dt][2];
      pb[8 + 3*HD + 16*dt + r] = vacc[dt][3];
    }
  }
}

// ---------------------------------------------------------------------------
// Combine partials across chunks, divide by softmax sum, apply sigmoid gate.
// One block per q head, thread = dim.
// ---------------------------------------------------------------------------
__global__ __launch_bounds__(128) void attn_combine(float* __restrict__ ws)
{
  const int h = blockIdx.x;
  const int d = threadIdx.x;
  const int g = h >> 2, i = h & 3;
  const float* pg = ws + WS_PART + (size_t)g * NCHUNK * PART_STRIDE;
  float M = -1e30f;
  for (int c = 0; c < NCHUNK; ++c) M = fmaxf(M, pg[(size_t)c*PART_STRIDE + i]);
  float L = 0.f, acc = 0.f;
  for (int c = 0; c < NCHUNK; ++c){
    const float* pb = pg + (size_t)c*PART_STRIDE;
    float f = __expf(pb[i] - M);
    L   += pb[4 + i] * f;
    acc += pb[8 + i*HD + d] * f;
  }
  float attn = acc / L;
  float gate = ws[WS_QG + h*(2*HD) + HD + d];
  float sig  = 1.0f / (1.0f + __expf(-gate));
  ws[WS_GATT + h*HD + d] = attn * sig;
}

// ---------------------------------------------------------------------------
extern "C" void kernel_launch(void* const* d_in, const int* in_sizes, int n_in,
                              void* d_out, int out_size, void* d_ws, size_t ws_size,
                              hipStream_t stream)
{
  (void)in_sizes; (void)n_in; (void)out_size; (void)ws_size;
  const float* hidden = (const float*)d_in[0];
  const float* pos    = (const float*)d_in[1];
  const float* cosb   = (const float*)d_in[2];
  const float* sinb   = (const float*)d_in[3];
  const float* kcache = (const float*)d_in[4];
  const float* vcache = (const float*)d_in[5];
  const float* qw     = (const float*)d_in[6];
  const float* kw     = (const float*)d_in[7];
  const float* vw     = (const float*)d_in[8];
  const float* ow     = (const float*)d_in[9];
  const float* qnw    = (const float*)d_in[10];
  const float* knw    = (const float*)d_in[11];

  float* out    = (float*)d_out;
  float* kc_out = out + H;                                   // new_k_cache
  float* vc_out = kc_out + (size_t)NKV*MAX_SEQ*HD;           // new_v_cache
  float* ws     = (float*)d_ws;
  const size_t cache_bytes = (size_t)NKV*MAX_SEQ*HD*sizeof(float);

  // bulk cache copy into output (then one row is overwritten)
  hipMemcpyAsync(kc_out, kcache, cache_bytes, hipMemcpyDeviceToDevice, stream);
  hipMemcpyAsync(vc_out, vcache, cache_bytes, hipMemcpyDeviceToDevice, stream);

  // QKV projections (GEMV, one wave per output row)
  gemv2048<<<512, 256, 0, stream>>>(qw, hidden, ws + WS_QG, NH*HD*2);
  gemv2048<<<64,  256, 0, stream>>>(kw, hidden, ws + WS_KP, NKV*HD);
  gemv2048<<<64,  256, 0, stream>>>(vw, hidden, ws + WS_VP, NKV*HD);

  // per-head rmsnorm + rope, write k/v rows into new caches
  norm_rope_write<<<NH + NKV, 128, 0, stream>>>(ws, qnw, knw, cosb, sinb, pos,
                                                kc_out, vc_out);

  // flash-decode partials with fp32 WMMA
  attn_partial<<<dim3(NKV, NCHUNK), 32, 0, stream>>>(ws + WS_QF, kc_out, vc_out,
                                                     pos, ws + WS_PART);

  // combine + sigmoid gating
  attn_combine<<<NH, 128, 0, stream>>>(ws);

  // output projection
  gemv2048<<<256, 256, 0, stream>>>(ow, ws + WS_GATT, out, H);
}